// GCN_37151467111211
// MI455X (gfx1250) — compile-verified
//
#include <hip/hip_runtime.h>

#define NFEAT 256
#define NHID 128
#define NCLASS 8

typedef __attribute__((ext_vector_type(2))) float v2f;
typedef __attribute__((ext_vector_type(8))) float v8f;

// ---------------------------------------------------------------- utilities
__global__ void zero_f32(float* __restrict__ p, long n) {
    long i = (long)blockIdx.x * blockDim.x + threadIdx.x;
    if (i < n) p[i] = 0.0f;
}

__global__ void degree_kernel(const int* __restrict__ src, const int* __restrict__ dst,
                              float* __restrict__ deg_out, float* __restrict__ deg_in,
                              int nedges) {
    int e = blockIdx.x * blockDim.x + threadIdx.x;
    if (e < nedges) {
        atomicAdd(&deg_out[src[e]], 1.0f);
        atomicAdd(&deg_in[dst[e]], 1.0f);
    }
}

// in-place: deg -> norm = deg>0 ? rsqrt(max(deg,1)) : 0
__global__ void norm_kernel(float* __restrict__ deg_out, float* __restrict__ deg_in, int n) {
    int i = blockIdx.x * blockDim.x + threadIdx.x;
    if (i < n) {
        float d0 = deg_out[i];
        deg_out[i] = (d0 > 0.0f) ? rsqrtf(fmaxf(d0, 1.0f)) : 0.0f;
        float d1 = deg_in[i];
        deg_in[i] = (d1 > 0.0f) ? rsqrtf(fmaxf(d1, 1.0f)) : 0.0f;
    }
}

// ---------------------------------------------------------------- GEMM1: h = (x @ W1) * norm_src
// Block = 256 thr (8 wave32), tile = 64 rows x 128 cols.
// Each wave owns one 16-col block and 4 row sub-tiles (4 accumulators),
// so each B (W1) fragment fetched from L2 feeds 4 V_WMMA_F32_16X16X4_F32.
#define MT 64   // rows per block
__global__ __launch_bounds__(256) void gemm1_wmma(
    const float* __restrict__ x,        // [n, NFEAT]
    const float* __restrict__ W1,       // [NFEAT, NHID]
    const float* __restrict__ norm_src, // [n]
    float* __restrict__ h,              // [n, NHID]
    int nnodes) {
    __shared__ float As[MT * NFEAT];    // 64 KB A tile

    const int rowBase = blockIdx.x * MT;
    const int tid  = threadIdx.x;       // 0..255
    const int wave = tid >> 5;          // 0..7 -> N column block
    const int lane = tid & 31;
    const int half = lane >> 4;         // 0: lanes 0-15, 1: lanes 16-31
    const int lid  = lane & 15;
    const bool fullTile = (rowBase + MT) <= nnodes;

    // stage A tile (64 rows x 256 K) into LDS as float4
    if (fullTile) {
        const float4* xv = (const float4*)(x + (size_t)rowBase * NFEAT);
        float4* asv = (float4*)As;
        for (int i = tid; i < MT * NFEAT / 4; i += 256) asv[i] = xv[i];
    } else {
        for (int i = tid; i < MT * NFEAT; i += 256) {
            int r = i / NFEAT, k = i % NFEAT;
            int row = rowBase + r;
            As[i] = (row < nnodes) ? x[(size_t)row * NFEAT + k] : 0.0f;
        }
    }
    __syncthreads();

    const int colBase = wave * 16;
    v8f acc0 = {0.f,0.f,0.f,0.f,0.f,0.f,0.f,0.f};
    v8f acc1 = acc0, acc2 = acc0, acc3 = acc0;

    for (int k0 = 0; k0 < NFEAT; k0 += 4) {
        const int ka = k0 + 2 * half;     // A/B K split by lane half
        // B 4x16 f32 frag (shared across the 4 row sub-tiles)
        v2f b;
        b.x = W1[(size_t)ka * NHID + colBase + lid];
        b.y = W1[(size_t)(ka + 1) * NHID + colBase + lid];
        // A 16x4 f32 frags from LDS, one per row sub-tile
        const float* a0 = &As[(0 * 16 + lid) * NFEAT + ka];
        const float* a1 = &As[(1 * 16 + lid) * NFEAT + ka];
        const float* a2 = &As[(2 * 16 + lid) * NFEAT + ka];
        const float* a3 = &As[(3 * 16 + lid) * NFEAT + ka];
        v2f f0; f0.x = a0[0]; f0.y = a0[1];
        v2f f1; f1.x = a1[0]; f1.y = a1[1];
        v2f f2; f2.x = a2[0]; f2.y = a2[1];
        v2f f3; f3.x = a3[0]; f3.y = a3[1];
        acc0 = __builtin_amdgcn_wmma_f32_16x16x4_f32(false, f0, false, b, (short)0, acc0, false, false);
        acc1 = __builtin_amdgcn_wmma_f32_16x16x4_f32(false, f1, false, b, (short)0, acc1, false, false);
        acc2 = __builtin_amdgcn_wmma_f32_16x16x4_f32(false, f2, false, b, (short)0, acc2, false, false);
        acc3 = __builtin_amdgcn_wmma_f32_16x16x4_f32(false, f3, false, b, (short)0, acc3, false, false);
    }

    // C/D layout: VGPR i -> row (i + 8*half), col = lid; scale by norm_src
    v8f accs[4] = {acc0, acc1, acc2, acc3};
    if (fullTile) {
        #pragma unroll
        for (int rt = 0; rt < 4; ++rt) {
            #pragma unroll
            for (int i = 0; i < 8; ++i) {
                int row = rowBase + rt * 16 + i + 8 * half;
                h[(size_t)row * NHID + colBase + lid] = accs[rt][i] * norm_src[row];
            }
        }
    } else {
        #pragma unroll
        for (int rt = 0; rt < 4; ++rt) {
            #pragma unroll
            for (int i = 0; i < 8; ++i) {
                int row = rowBase + rt * 16 + i + 8 * half;
                if (row < nnodes)
                    h[(size_t)row * NHID + colBase + lid] = accs[rt][i] * norm_src[row];
            }
        }
    }
}

// ---------------------------------------------------------------- edge scatter: agg[dst] += h[src]
// One wave32 per edge; 128 feats / 32 lanes = 1 float4 per lane.
__global__ __launch_bounds__(256) void edge_agg(
    const int* __restrict__ src, const int* __restrict__ dst,
    const float* __restrict__ h, float* __restrict__ agg, int nedges) {
    long gid = (long)blockIdx.x * blockDim.x + threadIdx.x;
    int e = (int)(gid >> 5);
    int lane = (int)(gid & 31);
    if (e >= nedges) return;
    int s = src[e];
    int d = dst[e];
    // hint-prefetch an upcoming source row (gfx1250 global_prefetch path)
    int ep = e + 64;
    if (ep < nedges && lane == 0) {
        __builtin_prefetch(&h[(size_t)src[ep] * NHID], 0, 1);
    }
    const float4* hs = (const float4*)(h + (size_t)s * NHID);
    float4 v = hs[lane];
    float* ad = agg + (size_t)d * NHID + lane * 4;
    atomicAdd(ad + 0, v.x);
    atomicAdd(ad + 1, v.y);
    atomicAdd(ad + 2, v.z);
    atomicAdd(ad + 3, v.w);
}

// ---------------------------------------------------------------- finalize: agg = relu(agg*norm_dst + b1)
__global__ void finalize_relu(float* __restrict__ agg, const float* __restrict__ norm_dst,
                              const float* __restrict__ b1, long ntot) {
    long i = (long)blockIdx.x * blockDim.x + threadIdx.x;
    if (i < ntot) {
        int node = (int)(i / NHID);
        int f = (int)(i % NHID);
        float v = agg[i] * norm_dst[node] + b1[f];
        agg[i] = v > 0.0f ? v : 0.0f;
    }
}

// ---------------------------------------------------------------- GEMM2: out = h @ W2 + b2 (NHID x NCLASS)
__global__ __launch_bounds__(256) void gemm2(
    const float* __restrict__ h, const float* __restrict__ W2,
    const float* __restrict__ b2, float* __restrict__ out, int nnodes) {
    __shared__ float W2s[NHID * NCLASS];  // 4 KB
    for (int i = threadIdx.x; i < NHID * NCLASS; i += 256) W2s[i] = W2[i];
    __syncthreads();
    long idx = (long)blockIdx.x * blockDim.x + threadIdx.x;
    int node = (int)(idx >> 3);
    int c = (int)(idx & 7);
    if (node >= nnodes) return;
    const float* hr = h + (size_t)node * NHID;
    float acc = b2[c];
    #pragma unroll 8
    for (int k = 0; k < NHID; ++k) acc += hr[k] * W2s[k * NCLASS + c];
    out[(size_t)node * NCLASS + c] = acc;
}

// ---------------------------------------------------------------- launch
extern "C" void kernel_launch(void* const* d_in, const int* in_sizes, int n_in,
                              void* d_out, int out_size, void* d_ws, size_t ws_size,
                              hipStream_t stream) {
    const float* x  = (const float*)d_in[0];
    const int*   sr = (const int*)d_in[1];
    const int*   ds = (const int*)d_in[2];
    const float* W1 = (const float*)d_in[3];
    const float* b1 = (const float*)d_in[4];
    const float* W2 = (const float*)d_in[5];
    const float* b2 = (const float*)d_in[6];
    float* out = (float*)d_out;

    const int nnodes = in_sizes[0] / NFEAT;
    const int nedges = in_sizes[1];

    // workspace layout (floats):
    //   [0, n)            deg_out -> norm_src
    //   [n, 2n)           deg_in  -> norm_dst
    //   [2n, 2n+n*128)    h  (projected + src-normalized)
    //   [.., +n*128)      agg
    float* ws       = (float*)d_ws;
    float* deg_out  = ws;
    float* deg_in   = ws + nnodes;
    float* h        = ws + 2 * (size_t)nnodes;
    float* agg      = h + (size_t)nnodes * NHID;

    const int B = 256;

    // degrees + norms
    long ndeg = 2L * nnodes;
    zero_f32<<<(int)((ndeg + B - 1) / B), B, 0, stream>>>(deg_out, ndeg);
    degree_kernel<<<(nedges + B - 1) / B, B, 0, stream>>>(sr, ds, deg_out, deg_in, nedges);
    norm_kernel<<<(nnodes + B - 1) / B, B, 0, stream>>>(deg_out, deg_in, nnodes);

    // GEMM1 (WMMA fp32): h = (x @ W1) * norm_src
    gemm1_wmma<<<(nnodes + MT - 1) / MT, B, 0, stream>>>(x, W1, deg_out, h, nnodes);

    // zero agg, edge scatter
    long naggf = (long)nnodes * NHID;
    zero_f32<<<(int)((naggf + B - 1) / B), B, 0, stream>>>(agg, naggf);
    long lanes = (long)nedges * 32;
    edge_agg<<<(int)((lanes + B - 1) / B), B, 0, stream>>>(sr, ds, h, agg, nedges);

    // finalize + relu (in place on agg)
    finalize_relu<<<(int)((naggf + B - 1) / B), B, 0, stream>>>(agg, deg_in, b1, naggf);

    // GEMM2 -> out
    long nout = (long)nnodes * NCLASS;
    gemm2<<<(int)((nout + B - 1) / B), B, 0, stream>>>(agg, W2, b2, out, nnodes);
}